// Base_64304250356210
// MI455X (gfx1250) — compile-verified
//
#include <hip/hip_runtime.h>
#include <math.h>

typedef __attribute__((ext_vector_type(2))) float v2f;
typedef __attribute__((ext_vector_type(8))) float v8f;

#define B_    128
#define L_    512
#define H_    128
#define TWOH  256
#define THREEH 384
#define FIVEH 640
#define R_    64

// meta per batch: 8 ints
// [0]=ci [1]=ip [2]=ipp [3]=inext [4]=action [5]=noact [6]=input_empty [7]=stack_empty

__global__ __launch_bounds__(128) void prep_kernel(
    const float* __restrict__ stack, const float* __restrict__ data,
    const float* __restrict__ cursors, const float* __restrict__ stack_mask,
    const float* __restrict__ mask,
    int* __restrict__ meta, float* __restrict__ Hmat, float* __restrict__ Cmat,
    float* __restrict__ inputc, float* __restrict__ Sin)
{
    int b = blockIdx.x;
    int t = threadIdx.x;
    __shared__ float red[3][128];
    float s0 = 0.f, s1 = 0.f, s2 = 0.f;
    for (int j = t; j < L_; j += 128) {
        s0 += mask[b * L_ + j];
        s1 += cursors[b * L_ + j];
        s2 += stack_mask[b * L_ + j];
    }
    red[0][t] = s0; red[1][t] = s1; red[2][t] = s2;
    __syncthreads();
    for (int off = 64; off > 0; off >>= 1) {
        if (t < off) {
            red[0][t] += red[0][t + off];
            red[1][t] += red[1][t + off];
            red[2][t] += red[2][t + off];
        }
        __syncthreads();
    }
    __shared__ int mi[3];
    if (t == 0) {
        float sl = red[0][0], cp = red[1][0], ss = red[2][0];
        int ci   = (int)(cp + 0.5f) - 1;
        int issi = (int)(ss + 0.5f);
        int ip   = issi - 2;
        int ipp  = issi - 3;
        int inxt = issi < (L_ - 1) ? issi : (L_ - 1);
        int ie   = ((sl - cp) == -1.0f) ? 1 : 0;
        int se   = (ss <= 1.0f) ? 1 : 0;
        int* mb = meta + b * 8;
        mb[0] = ci; mb[1] = ip; mb[2] = ipp; mb[3] = inxt; mb[6] = ie; mb[7] = se;
        mi[0] = ci; mi[1] = ip; mi[2] = ipp;
    }
    __syncthreads();
    int ci = mi[0], ip = mi[1], ipp = mi[2];
    const float* sp  = (ip  >= 0) ? stack + ((size_t)b * L_ + ip)  * TWOH : nullptr;
    const float* spp = (ipp >= 0) ? stack + ((size_t)b * L_ + ipp) * TWOH : nullptr;
    const float* dc  = (ci  >= 0) ? data  + ((size_t)b * L_ + ci)  * TWOH : nullptr;

    float hv0 = sp  ? sp [H_ + t] : 0.f;   // prev h-half
    float hv1 = spp ? spp[H_ + t] : 0.f;   // prev_prev h-half
    float cv0 = sp  ? sp [t]      : 0.f;   // prev c-half
    float cv1 = spp ? spp[t]      : 0.f;   // prev_prev c-half
    float d0  = dc  ? dc [t]      : 0.f;   // input_current c-half
    float d1  = dc  ? dc [H_ + t] : 0.f;   // input_current h-half

    Hmat[b * TWOH + t]       = hv0;
    Hmat[b * TWOH + H_ + t]  = hv1;
    Cmat[b * TWOH + t]       = cv0;
    Cmat[b * TWOH + H_ + t]  = cv1;
    inputc[b * TWOH + t]      = d0;
    inputc[b * TWOH + H_ + t] = d1;
    Sin[b * THREEH + t]           = hv0;
    Sin[b * THREEH + H_ + t]      = hv1;
    Sin[b * THREEH + 2 * H_ + t]  = d1;
}

// C[M,N] = A[M,K] @ Bm[K,N] + bias[N], row-major, one wave per 16x16 tile.
// Uses V_WMMA_F32_16X16X4_F32 (full fp32 precision).
__global__ __launch_bounds__(32) void wmma_gemm_f32(
    const float* __restrict__ A, const float* __restrict__ Bm,
    const float* __restrict__ bias, float* __restrict__ C, int K, int N)
{
    int nb   = blockIdx.x * 16;
    int mb   = blockIdx.y * 16;
    int lane = threadIdx.x;
    int half = lane >> 4;     // 0: K,K+1 ; 1: K+2,K+3
    int l15  = lane & 15;

    v8f acc = {};
    for (int k = 0; k < K; k += 4) {
        int ka = k + half * 2;
        v2f a, bv;
        // A 16x4: lane (m=l15) holds K=ka, ka+1
        a.x  = A[(size_t)(mb + l15) * K + ka + 0];
        a.y  = A[(size_t)(mb + l15) * K + ka + 1];
        // B 4x16: lane (n=l15) holds K=ka, ka+1
        bv.x = Bm[(size_t)(ka + 0) * N + nb + l15];
        bv.y = Bm[(size_t)(ka + 1) * N + nb + l15];
        acc = __builtin_amdgcn_wmma_f32_16x16x4_f32(
            false, a, false, bv, (short)0, acc, false, false);
    }
    // C/D layout: VGPR r -> M = mb + half*8 + r, N = nb + l15
    int m0 = mb + half * 8;
    int n  = nb + l15;
    float bvv = bias ? bias[n] : 0.0f;
#pragma unroll
    for (int r = 0; r < 8; ++r) {
        C[(size_t)(m0 + r) * N + n] = acc[r] + bvv;
    }
}

__global__ __launch_bounds__(128) void act_kernel(
    const float* __restrict__ Q, const float* __restrict__ S,
    const float* __restrict__ Cmat,
    const float* __restrict__ W_S2, const float* __restrict__ b_S2,
    int* __restrict__ meta, float* __restrict__ reduced,
    float* __restrict__ policy_out)
{
    int b = blockIdx.x;
    int t = threadIdx.x;
    __shared__ float sld[R_];
    if (t < R_) sld[t] = fmaxf(S[b * R_ + t], 0.0f);
    __syncthreads();
    if (t == 0) {
        float p0 = b_S2[0], p1 = b_S2[1];
        for (int k = 0; k < R_; ++k) {
            p0 += sld[k] * W_S2[k * 2 + 0];
            p1 += sld[k] * W_S2[k * 2 + 1];
        }
        p0 = expf(p0); p1 = expf(p1);
        policy_out[b * 2 + 0] = p0;
        policy_out[b * 2 + 1] = p1;
        int act = (p0 >= p1) ? 1 : 0;
        int* mb = meta + b * 8;
        int ie = mb[6], se = mb[7];
        if (ie) act = 0;          // input empty -> reduce
        if (se) act = 1;          // stack empty -> shift (overrides)
        mb[4] = act;
        mb[5] = (ie && se) ? 1 : 0;
    }
    // tree-LSTM gated reduce, H lanes per batch
    const float* q = Q + (size_t)b * FIVEH;
    float q0 = q[t], q1 = q[H_ + t], q2 = q[2 * H_ + t], q3 = q[3 * H_ + t];
    float qt = q[4 * H_ + t];
    float g0 = 1.0f / (1.0f + expf(-q0));
    float g1 = 1.0f / (1.0f + expf(-q1));
    float g2 = 1.0f / (1.0f + expf(-q2));
    float g3 = 1.0f / (1.0f + expf(-q3));
    float tq = tanhf(qt);
    float c  = g1 * Cmat[b * TWOH + t] + g2 * Cmat[b * TWOH + H_ + t] + g0 * tq;
    float hh = g3 * c;
    reduced[b * TWOH + t]      = c;
    reduced[b * TWOH + H_ + t] = hh;
}

// Bandwidth phase: copy 67MB stack -> out, patching <=2 rows per batch element.
// 256 threads / block, 4 rows per block, one float4 per thread.
__global__ __launch_bounds__(256) void scatter_kernel(
    const float* __restrict__ stack, const float* __restrict__ inputc,
    const float* __restrict__ reduced, const int* __restrict__ meta,
    float* __restrict__ out)
{
    int t    = threadIdx.x;
    int row  = blockIdx.x * 4 + (t >> 6);   // b*L + l
    int col4 = t & 63;                      // float4 index within the 256-f32 row
    int b = row >> 9;                       // /L_
    int l = row & (L_ - 1);

    const int* mb = meta + b * 8;
    int ip = mb[1], ipp = mb[2], inxt = mb[3], act = mb[4], noact = mb[5];

    const float4* src = reinterpret_cast<const float4*>(stack + (size_t)row * TWOH);
    float4 v = src[col4];
    if (!noact) {
        if (act == 1) {
            if (l == inxt)
                v = reinterpret_cast<const float4*>(inputc + (size_t)b * TWOH)[col4];
        } else {
            if (l == ip)
                v = make_float4(0.f, 0.f, 0.f, 0.f);
            else if (l == ipp)   // never taken when ipp == -1 (empty-mask no-op)
                v = reinterpret_cast<const float4*>(reduced + (size_t)b * TWOH)[col4];
        }
    }
    reinterpret_cast<float4*>(out + (size_t)row * TWOH)[col4] = v;
}

extern "C" void kernel_launch(void* const* d_in, const int* in_sizes, int n_in,
                              void* d_out, int out_size, void* d_ws, size_t ws_size,
                              hipStream_t stream) {
    const float* stack      = (const float*)d_in[0];
    const float* data       = (const float*)d_in[1];
    const float* cursors    = (const float*)d_in[2];
    const float* stack_mask = (const float*)d_in[3];
    const float* mask       = (const float*)d_in[4];
    const float* W_R        = (const float*)d_in[5];
    const float* b_R        = (const float*)d_in[6];
    const float* W_S1       = (const float*)d_in[7];
    const float* b_S1       = (const float*)d_in[8];
    const float* W_S2       = (const float*)d_in[9];
    const float* b_S2       = (const float*)d_in[10];

    char* ws = (char*)d_ws;
    size_t off = 0;
    int*   meta    = (int*)(ws + off);   off += 4096;                       // 128*8*4
    float* Hmat    = (float*)(ws + off); off += (size_t)B_ * TWOH * 4;      // 131072
    float* Cmat    = (float*)(ws + off); off += (size_t)B_ * TWOH * 4;
    float* inputc  = (float*)(ws + off); off += (size_t)B_ * TWOH * 4;
    float* Sin     = (float*)(ws + off); off += (size_t)B_ * THREEH * 4;
    float* Q       = (float*)(ws + off); off += (size_t)B_ * FIVEH * 4;
    float* S       = (float*)(ws + off); off += (size_t)B_ * R_ * 4;
    float* reduced = (float*)(ws + off); off += (size_t)B_ * TWOH * 4;

    float* out_stack  = (float*)d_out;
    float* out_policy = out_stack + (size_t)B_ * L_ * TWOH;

    prep_kernel<<<B_, 128, 0, stream>>>(stack, data, cursors, stack_mask, mask,
                                        meta, Hmat, Cmat, inputc, Sin);

    // Q[128,640] = Hmat[128,256] @ W_R + b_R
    wmma_gemm_f32<<<dim3(FIVEH / 16, B_ / 16), 32, 0, stream>>>(
        Hmat, W_R, b_R, Q, TWOH, FIVEH);

    // S[128,64] = Sin[128,384] @ W_S1 + b_S1
    wmma_gemm_f32<<<dim3(R_ / 16, B_ / 16), 32, 0, stream>>>(
        Sin, W_S1, b_S1, S, THREEH, R_);

    act_kernel<<<B_, 128, 0, stream>>>(Q, S, Cmat, W_S2, b_S2,
                                       meta, reduced, out_policy);

    scatter_kernel<<<(B_ * L_) / 4, 256, 0, stream>>>(stack, inputc, reduced,
                                                      meta, out_stack);
}